// SecondaryCapsule_38646115729540
// MI455X (gfx1250) — compile-verified
//
#include <hip/hip_runtime.h>

typedef __attribute__((ext_vector_type(2))) float v2f;
typedef __attribute__((ext_vector_type(8))) float v8f;

#define BATCH 64
#define N1    1152
#define PD    8
#define N2    128
#define DD    16
#define KDIM  (N1 * PD)   // 9216
#define JDIM  (N2 * DD)   // 2048

// ---------------------------------------------------------------------------
// Kernel 1: s0[b][j] = sum_k A[b][k] * Bm[k][j]
//   A[b][k]  = x[b*9216 + k]                      (k = i*8 + p)
//   Bm[k][j] = W[(k>>3)*16384 + j*8 + (k&7)]      (j = n*16 + d)
// One wave per 16x16 C tile, V_WMMA_F32_16X16X4_F32 over K.
// ---------------------------------------------------------------------------
__global__ __launch_bounds__(128) void gemm_s0_kernel(
    const float* __restrict__ x, const float* __restrict__ W,
    float* __restrict__ s0)
{
    const int wave = blockIdx.x * 4 + (threadIdx.x >> 5);   // 0..511
    const int lane = threadIdx.x & 31;
    const int half = lane >> 4;        // 0: K pair {0,1}, 1: K pair {2,3}
    const int ml   = lane & 15;
    const int m0   = (wave & 3) * 16;  // 4 m-tiles (B=64)
    const int n0   = (wave >> 2) * 16; // 128 n-tiles (JDIM=2048)
    const int j    = n0 + ml;

    const float* ap = x + (size_t)(m0 + ml) * KDIM + half * 2;
    v8f acc = {};

#pragma unroll 4
    for (int k0 = 0; k0 < KDIM; k0 += 4) {
        const int k = k0 + half * 2;
        // A frag: lane half 0 -> K=k0,k0+1 ; half 1 -> K=k0+2,k0+3
        float2 av = *(const float2*)(ap + k0);
        // B frag: VGPR0 holds K=k0(+2*half), VGPR1 holds K=k0+1(+2*half), col j
        // (k&7) is always even here, so the pair never crosses the P=8 boundary.
        float2 bv = *(const float2*)(W + (size_t)(k >> 3) * (N2 * DD * PD)
                                       + (size_t)j * PD + (k & 7));
        v2f a; a.x = av.x; a.y = av.y;
        v2f b; b.x = bv.x; b.y = bv.y;
        acc = __builtin_amdgcn_wmma_f32_16x16x4_f32(
            false, a, false, b, (short)0, acc, false, false);
    }

    // C layout: VGPR v -> row m0+v (lanes 0-15) / m0+v+8 (lanes 16-31), col j
#pragma unroll
    for (int v = 0; v < 8; ++v)
        s0[(size_t)(m0 + v + half * 8) * JDIM + j] = acc[v];
}

// ---------------------------------------------------------------------------
// Squash: one thread per (b,n) capsule of 16 dims. out = sq/(1+sq) * s/norm
// ---------------------------------------------------------------------------
__global__ __launch_bounds__(256) void squash_kernel(
    const float* __restrict__ in, float* __restrict__ out, float scale)
{
    const int idx = blockIdx.x * 256 + threadIdx.x;   // 0..8191 = b*128+n
    const float4* sp = (const float4*)(in + (size_t)idx * DD);
    float4 r[4];
    float sq = 0.f;
#pragma unroll
    for (int q = 0; q < 4; ++q) {
        float4 t = sp[q];
        t.x *= scale; t.y *= scale; t.z *= scale; t.w *= scale;
        sq += t.x * t.x + t.y * t.y + t.z * t.z + t.w * t.w;
        r[q] = t;
    }
    const float norm = sqrtf(sq + 1.1920929e-07f);     // FLT_EPSILON
    const float f = (sq / (1.f + sq)) / norm;
    float4* op = (float4*)(out + (size_t)idx * DD);
#pragma unroll
    for (int q = 0; q < 4; ++q) {
        float4 t = r[q];
        t.x *= f; t.y *= f; t.z *= f; t.w *= f;
        op[q] = t;
    }
}

// ---------------------------------------------------------------------------
// Kernel 3: fused routing pass (recomputes pred from L2-resident W).
// Block = (BCH batches) x (IPB input capsules). Thread t owns capsule n=t>>1,
// d-half dh=(t&1)*8: its 64 W floats are contiguous and loaded once per i,
// reused across the BCH batches. s1 partials live in registers; one atomic
// flush per thread at the end.
// ---------------------------------------------------------------------------
#define IPB 64
#define BCH 4

__global__ __launch_bounds__(256) void routing_kernel(
    const float* __restrict__ x, const float* __restrict__ W,
    const float* __restrict__ v0, float* __restrict__ s1)
{
    __shared__ float sh_x[BCH * IPB * PD];   // 8 KB
    __shared__ float sh_v0[BCH * JDIM];      // 32 KB
    __shared__ float sh_raw[N2];
    __shared__ float sh_c[N2];
    __shared__ float sh_rmax[8];
    __shared__ float sh_rsum[8];

    const int t    = threadIdx.x;
    const int i0   = blockIdx.x * IPB;
    const int b0   = blockIdx.y * BCH;
    const int n    = t >> 1;
    const int dh   = (t & 1) * 8;
    const int wid  = t >> 5;
    const int lane = t & 31;

    for (int u = t; u < BCH * IPB * PD; u += 256) {
        const int bb  = u / (IPB * PD);
        const int rem = u - bb * (IPB * PD);
        sh_x[u] = x[(size_t)(b0 + bb) * (N1 * PD) + (size_t)i0 * PD + rem];
    }
    for (int u = t; u < BCH * JDIM; u += 256) {
        const int bb = u >> 11;
        sh_v0[u] = v0[(size_t)(b0 + bb) * JDIM + (u & (JDIM - 1))];
    }
    __syncthreads();

    float acc[BCH][8];
#pragma unroll
    for (int bb = 0; bb < BCH; ++bb)
#pragma unroll
        for (int d = 0; d < 8; ++d) acc[bb][d] = 0.f;

    for (int ii = 0; ii < IPB; ++ii) {
        // Thread's 64 contiguous W floats: W[i][n][dh..dh+7][0..7]
        const float4* wp = (const float4*)(W
            + (size_t)(i0 + ii) * (N2 * DD * PD)
            + (size_t)n * (DD * PD) + (size_t)dh * PD);
        float4 wreg[16];
#pragma unroll
        for (int q = 0; q < 16; ++q) wreg[q] = wp[q];

#pragma unroll
        for (int bb = 0; bb < BCH; ++bb) {
            const float* xp = &sh_x[(bb * IPB + ii) * PD];
            float pred[8];
#pragma unroll
            for (int d = 0; d < 8; ++d) {
                const float4 w0 = wreg[d * 2], w1 = wreg[d * 2 + 1];
                pred[d] = w0.x * xp[0] + w0.y * xp[1] + w0.z * xp[2] + w0.w * xp[3]
                        + w1.x * xp[4] + w1.y * xp[5] + w1.z * xp[6] + w1.w * xp[7];
            }
            // agreement partial over this thread's 8 dims, combine with partner
            const float* vp = &sh_v0[bb * JDIM + n * DD + dh];
            float rp = 0.f;
#pragma unroll
            for (int d = 0; d < 8; ++d) rp += pred[d] * vp[d];
            rp += __shfl_xor(rp, 1, 32);
            if ((t & 1) == 0) sh_raw[n] = rp;
            __syncthreads();

            // block softmax over the 128 output capsules
            float m = (t < N2) ? sh_raw[t] : -3.4e38f;
#pragma unroll
            for (int off = 16; off >= 1; off >>= 1)
                m = fmaxf(m, __shfl_xor(m, off, 32));
            if (lane == 0) sh_rmax[wid] = m;
            __syncthreads();
            float gmax = sh_rmax[0];
#pragma unroll
            for (int q = 1; q < 8; ++q) gmax = fmaxf(gmax, sh_rmax[q]);

            float e = 0.f;
            if (t < N2) { e = __expf(sh_raw[t] - gmax); sh_c[t] = e; }
            float ssum = e;
#pragma unroll
            for (int off = 16; off >= 1; off >>= 1)
                ssum += __shfl_xor(ssum, off, 32);
            if (lane == 0) sh_rsum[wid] = ssum;
            __syncthreads();
            float gsum = 0.f;
#pragma unroll
            for (int q = 0; q < 8; ++q) gsum += sh_rsum[q];

            const float cn = sh_c[n] / gsum;
#pragma unroll
            for (int d = 0; d < 8; ++d) acc[bb][d] += cn * pred[d];
            __syncthreads();   // protect sh_raw/sh_c for next iteration
        }
    }

    float* sp = s1 + (size_t)b0 * JDIM + n * DD + dh;
#pragma unroll
    for (int bb = 0; bb < BCH; ++bb)
#pragma unroll
        for (int d = 0; d < 8; ++d)
            atomicAdd(sp + (size_t)bb * JDIM + d, acc[bb][d]);
}

// ---------------------------------------------------------------------------
extern "C" void kernel_launch(void* const* d_in, const int* in_sizes, int n_in,
                              void* d_out, int out_size, void* d_ws, size_t ws_size,
                              hipStream_t stream) {
    const float* x = (const float*)d_in[0];   // [64,1152,8]
    const float* W = (const float*)d_in[1];   // [1152,128,16,8]
    float* out = (float*)d_out;               // [64,128,16]

    float* s_buf = (float*)d_ws;              // s0, later reused as s1 (512 KB)
    float* v0    = s_buf + (size_t)BATCH * JDIM;  // v0 (512 KB)

    // Pass 1: s0 = x_flat @ W_flat   (WMMA f32 16x16x4)
    gemm_s0_kernel<<<128, 128, 0, stream>>>(x, W, s_buf);
    // v0 = squash(s0 / 128)
    squash_kernel<<<(BATCH * N2) / 256, 256, 0, stream>>>(s_buf, v0, 1.f / 128.f);
    // reuse s_buf as s1 accumulator
    hipMemsetAsync(s_buf, 0, (size_t)BATCH * JDIM * sizeof(float), stream);
    // fused agreement + softmax + weighted sum (pred recomputed from L2)
    routing_kernel<<<dim3(N1 / IPB, BATCH / BCH), 256, 0, stream>>>(x, W, v0, s_buf);
    // out = squash(s1)
    squash_kernel<<<(BATCH * N2) / 256, 256, 0, stream>>>(s_buf, out, 1.f);
}